// TreeNodeClassifier_34359738368489
// MI455X (gfx1250) — compile-verified
//
#include <hip/hip_runtime.h>
#include <hip/hip_bf16.h>

typedef float v2f __attribute__((ext_vector_type(2)));
typedef float v8f __attribute__((ext_vector_type(8)));

// Hardware FP32 atomic add (global_atomic_add_f32, no CAS loop).
__device__ __forceinline__ void hw_atomic_add(float* p, float v) {
#if defined(__HIP_DEVICE_COMPILE__) && defined(__AMDGCN__)
    unsafeAtomicAdd(p, v);
#else
    atomicAdd(p, v);
#endif
}

// ---------------------------------------------------------------------------
// float4 copy (n4 = number of float4 elements)
// ---------------------------------------------------------------------------
__global__ __launch_bounds__(256)
void copy_f32x4(const float4* __restrict__ in, float4* __restrict__ out, long n4) {
    long i = (long)blockIdx.x * blockDim.x + threadIdx.x;
    if (i < n4) out[i] = in[i];
}

// ---------------------------------------------------------------------------
// Scatter-add: agg[dst[e]] += h[src[e]]  (F floats per edge, F % 4 == 0)
// One thread per (edge, quad-of-4-floats); consecutive threads cover
// consecutive quads of the same row -> coalesced 128B/256B row reads.
// ---------------------------------------------------------------------------
__global__ __launch_bounds__(256)
void scatter_add_f32(const float* __restrict__ h, const int* __restrict__ src,
                     const int* __restrict__ dst, float* __restrict__ agg,
                     int E, int F) {
    const int quads = F >> 2;
    long t = (long)blockIdx.x * blockDim.x + threadIdx.x;
    int e = (int)(t / quads);
    int q = (int)(t - (long)e * quads);
    if (e >= E) return;
    const int s = src[e];
    const int d = dst[e];
    const float4 v = reinterpret_cast<const float4*>(h + (long)s * F)[q];
    float* o = agg + (long)d * F + (q << 2);
    hw_atomic_add(o + 0, v.x);
    hw_atomic_add(o + 1, v.y);
    hw_atomic_add(o + 2, v.z);
    hw_atomic_add(o + 3, v.w);
}

// ---------------------------------------------------------------------------
// WMMA GEMM: C[M x N] = act(A[M x K] @ W[K x N] + bias[N])
//   - f32 everywhere, uses V_WMMA_F32_16X16X4_F32
//   - 8 waves / block, each wave computes one 16-row M-tile across all N-tiles
//   - requires M % 16 == 0, K % 4 == 0, N % 16 == 0 (true here: M=100000)
//
// ISA 7.12.2 layouts (wave32):
//   A 16x4 f32 : lanes 0-15 -> rows, VGPR0=K0, VGPR1=K1 ; lanes 16-31 -> K2,K3
//   B 4x16 f32 : lanes 0-15 -> cols, VGPR0=K0, VGPR1=K1 ; lanes 16-31 -> K2,K3
//   C 16x16 f32: VGPR r: lanes 0-15 -> row r, col=lane ; lanes 16-31 -> row r+8
// ---------------------------------------------------------------------------
__global__ __launch_bounds__(256)
void gemm_bias_act_wmma(const float* __restrict__ A, const float* __restrict__ W,
                        const float* __restrict__ bias, float* __restrict__ C,
                        int M, int K, int N, int do_relu) {
    const int lane  = threadIdx.x & 31;
    const int wave  = threadIdx.x >> 5;
    const int mtile = blockIdx.x * 8 + wave;
    if (mtile * 16 >= M) return;            // wave-uniform: EXEC stays all-1s

    const int half = lane >> 4;             // 0: lanes 0-15, 1: lanes 16-31
    const int l16  = lane & 15;
    const int row0 = mtile * 16;
    const int arow = row0 + l16;            // this lane's A row
    const int nTiles = N >> 4;

    for (int nt = 0; nt < nTiles; ++nt) {
        const int col = nt * 16 + l16;      // this lane's B/C column
        v8f acc = {};
        for (int k0 = 0; k0 < K; k0 += 4) {
            const int ka = k0 + (half << 1);
            v2f a, b;
            a.x = A[(long)arow * K + ka];
            a.y = A[(long)arow * K + ka + 1];
            b.x = W[(long)(ka    ) * N + col];
            b.y = W[(long)(ka + 1) * N + col];
            acc = __builtin_amdgcn_wmma_f32_16x16x4_f32(
                false, a, false, b, (short)0, acc, false, false);
        }
        const float bv  = bias[col];
        const int crow0 = row0 + (half << 3);
#pragma unroll
        for (int r = 0; r < 8; ++r) {
            float v = acc[r] + bv;
            if (do_relu) v = fmaxf(v, 0.0f);
            C[(long)(crow0 + r) * N + nt * 16 + l16] = v;
        }
    }
}

// ---------------------------------------------------------------------------
// Host-side launch sequence
// ---------------------------------------------------------------------------
extern "C" void kernel_launch(void* const* d_in, const int* in_sizes, int n_in,
                              void* d_out, int out_size, void* d_ws, size_t ws_size,
                              hipStream_t stream) {
    const float* x    = (const float*)d_in[0];
    const int*   ei   = (const int*)  d_in[1];
    const float* w1_0 = (const float*)d_in[2];
    const float* b1_0 = (const float*)d_in[3];
    const float* w2_0 = (const float*)d_in[4];
    const float* b2_0 = (const float*)d_in[5];
    const float* w1_1 = (const float*)d_in[6];
    const float* b1_1 = (const float*)d_in[7];
    const float* w2_1 = (const float*)d_in[8];
    const float* b2_1 = (const float*)d_in[9];
    const float* cw1  = (const float*)d_in[10];
    const float* cb1  = (const float*)d_in[11];
    const float* cw2  = (const float*)d_in[12];
    const float* cb2  = (const float*)d_in[13];

    const int Nn = in_sizes[0] / 32;        // 100000 nodes
    const int E  = in_sizes[1] / 2;         // 1600000 edges
    const int* src = ei;
    const int* dst = ei + E;

    float* buf0 = (float*)d_ws;             // N x 64
    float* buf1 = buf0 + (long)Nn * 64;     // N x 64

    const int mBlocks = ((Nn / 16) + 7) / 8;   // 8 waves (16-row tiles) per block

    // ---- TreeConv layer 0 (F = 32) ----
    {
        long n4 = (long)Nn * 8;                         // N*32 floats / 4
        copy_f32x4<<<(int)((n4 + 255) / 256), 256, 0, stream>>>(
            (const float4*)x, (float4*)buf0, n4);
        long thr = (long)E * 8;                         // E * (32/4)
        scatter_add_f32<<<(int)((thr + 255) / 256), 256, 0, stream>>>(
            x, src, dst, buf0, E, 32);
        gemm_bias_act_wmma<<<mBlocks, 256, 0, stream>>>(buf0, w1_0, b1_0, buf1, Nn, 32, 64, 1);
        gemm_bias_act_wmma<<<mBlocks, 256, 0, stream>>>(buf1, w2_0, b2_0, buf0, Nn, 64, 64, 0);
    }

    // ---- TreeConv layer 1 (F = 64), h0 currently in buf0 ----
    {
        long n4 = (long)Nn * 16;                        // N*64 floats / 4
        copy_f32x4<<<(int)((n4 + 255) / 256), 256, 0, stream>>>(
            (const float4*)buf0, (float4*)buf1, n4);
        long thr = (long)E * 16;                        // E * (64/4)
        scatter_add_f32<<<(int)((thr + 255) / 256), 256, 0, stream>>>(
            buf0, src, dst, buf1, E, 64);
        gemm_bias_act_wmma<<<mBlocks, 256, 0, stream>>>(buf1, w1_1, b1_1, buf0, Nn, 64, 64, 1);
        gemm_bias_act_wmma<<<mBlocks, 256, 0, stream>>>(buf0, w2_1, b2_1, buf1, Nn, 64, 64, 0);
    }

    // ---- classifier head, h1 currently in buf1 ----
    gemm_bias_act_wmma<<<mBlocks, 256, 0, stream>>>(buf1, cw1, cb1, buf0, Nn, 64, 64, 1);
    gemm_bias_act_wmma<<<mBlocks, 256, 0, stream>>>(buf0, cw2, cb2, (float*)d_out, Nn, 64, 16, 0);
}